// Injector_multimodal_32272384262584
// MI455X (gfx1250) — compile-verified
//
#include <hip/hip_runtime.h>
#include <stdint.h>

// ---------------- problem constants (from setup_inputs) ----------------
#define B_      2
#define LQ      4096
#define C_      384           // = HH*DH
#define HH      6
#define DH      64
#define PP      4
#define MM      4
#define GW      64            // spatial W
#define GH      64            // spatial H
#define R_      (B_*LQ)       // 8192 rows
#define NOFF    48            // H*Lv*P*2
#define NAW     24            // H*Lv*P

typedef __attribute__((ext_vector_type(2))) float v2f;
typedef __attribute__((ext_vector_type(8))) float v8f;

// ---------------- LayerNorm: one block (128 thr) per row of 384 ----------------
__global__ __launch_bounds__(128)
void ln_kernel(const float* __restrict__ src, const float* __restrict__ g,
               const float* __restrict__ b, float* __restrict__ dst) {
  const int row = blockIdx.x;
  const int t   = threadIdx.x;
  const float* x = src + (size_t)row * C_;
  float*       y = dst + (size_t)row * C_;
  float v0 = x[t], v1 = x[t + 128], v2 = x[t + 256];
  __shared__ float sh_s[128], sh_q[128];
  sh_s[t] = v0 + v1 + v2;
  sh_q[t] = v0*v0 + v1*v1 + v2*v2;
  __syncthreads();
  for (int o = 64; o > 0; o >>= 1) {
    if (t < o) { sh_s[t] += sh_s[t + o]; sh_q[t] += sh_q[t + o]; }
    __syncthreads();
  }
  const float mean = sh_s[0] * (1.0f / C_);
  const float var  = sh_q[0] * (1.0f / C_) - mean * mean;
  const float inv  = rsqrtf(var + 1e-6f);
  y[t]       = (v0 - mean) * inv * g[t]       + b[t];
  y[t + 128] = (v1 - mean) * inv * g[t + 128] + b[t + 128];
  y[t + 256] = (v2 - mean) * inv * g[t + 256] + b[t + 256];
}

// ---------------- out = query (vectorized copy) ----------------
__global__ __launch_bounds__(256)
void copy_kernel(const float4* __restrict__ s, float4* __restrict__ d, int n) {
  int i = blockIdx.x * blockDim.x + threadIdx.x;
  if (i < n) d[i] = s[i];
}

// ---------------- fp32 WMMA GEMM with async B-strip staging in LDS -----------
// Block = 256 thr (8 waves). Macro-tile: 128 rows x 16 cols. K = C_ = 384.
// The whole W[:, n0:n0+16] strip (384x16 f32 = 24KB) is async-copied to LDS
// once per block (GLOBAL_LOAD_ASYNC_TO_LDS_B128, ASYNCcnt), then each wave
// runs the 16x16x4 f32 WMMA K-loop with B fragments served from LDS.
// ACCUM=false: D[row,col]   = acc + bias[col]
// ACCUM=true : out[row,col] += gamma[col] * (acc + bias[col])
template <bool ACCUM>
__global__ __launch_bounds__(256)
void gemm_wmma_f32(const float* __restrict__ A, const float* __restrict__ W,
                   const float* __restrict__ bias, float* __restrict__ D,
                   const float* __restrict__ gamma, float* __restrict__ out) {
  __shared__ float Bs[C_ * 16];              // 24 KB, Bs[k*16 + c], LDS offset 0

  const int tid  = threadIdx.x;
  const int n0   = blockIdx.y * 16;
  const int m0   = blockIdx.x * 128 + (tid >> 5) * 16;   // wave-private row tile

  // ---- async stage: 1536 x b128 transfers, 6 per thread ----
  {
    #pragma unroll
    for (int it = 0; it < 6; ++it) {
      const int e = it * 256 + tid;          // e in [0,1536)
      const int k = e >> 2;                  // B row (0..383)
      const int q = e & 3;                   // 4-float quarter of the 16-col strip
      const uint32_t ldsoff = (uint32_t)((k * 16 + q * 4) * 4);   // bytes, 16B aligned
      const uint64_t gaddr  = (uint64_t)(uintptr_t)(W + (size_t)k * C_ + n0 + q * 4);
      asm volatile("global_load_async_to_lds_b128 %0, %1, off"
                   :: "v"(ldsoff), "v"(gaddr) : "memory");
    }
    asm volatile("s_wait_asynccnt 0" ::: "memory");   // my wave's copies done
  }
  __syncthreads();                                     // all waves' copies done

  const int lane = tid & 31;                 // wave32, EXEC all ones
  const int half = lane >> 4;                // 0: K0/K1 ; 1: K2/K3
  const int l16  = lane & 15;
  const int arow = m0 + l16;                 // A: row striped in lane, K in vgprs
  const int bcol = n0 + l16;                 // B/C/D: col (N) striped across lanes

  const float* Ap  = A  + (size_t)arow * C_ + half * 2;   // 8B aligned
  const float* Bsp = Bs + (half * 2) * 16 + l16;

  v8f c = {0.f, 0.f, 0.f, 0.f, 0.f, 0.f, 0.f, 0.f};
  #pragma unroll 4
  for (int k = 0; k < C_; k += 4) {
    v2f a = *(const v2f*)(Ap + k);           // A[arow][k+2*half .. +1]
    v2f bf;
    bf.x = Bsp[k * 16];                      // W[k+2*half  ][bcol] (from LDS)
    bf.y = Bsp[(k + 1) * 16];                // W[k+2*half+1][bcol]
    __builtin_prefetch(Ap + k + 64, 0, 0);   // global_prefetch_b8
    // 8 args: (neg_a, A, neg_b, B, c_mod, C, reuse_a, reuse_b)
    c = __builtin_amdgcn_wmma_f32_16x16x4_f32(
        false, a, false, bf, (short)0, c, false, false);
  }

  // ---- straight-line epilogue (templated, no runtime branch) ----
  const float bcv = bias[bcol];
  if (ACCUM) {
    const float gcv = gamma[bcol];
    float* op = out + (size_t)(m0 + half * 8) * C_ + bcol;
    #pragma unroll
    for (int r = 0; r < 8; ++r)
      op[(size_t)r * C_] += gcv * (c[r] + bcv);
  } else {
    float* dp = D + (size_t)(m0 + half * 8) * C_ + bcol;
    #pragma unroll
    for (int r = 0; r < 8; ++r)
      dp[(size_t)r * C_] = c[r] + bcv;
  }
}

// ---------------- sampling-offset / attention-weight projection ----------------
// one block (96 thr) per row: cols 0..47 -> off, 48..71 -> aw (pre-softmax)
__global__ __launch_bounds__(96)
void offaw_kernel(const float* __restrict__ qn,
                  const float* __restrict__ Woff, const float* __restrict__ boff,
                  const float* __restrict__ Waw,  const float* __restrict__ baw,
                  float* __restrict__ off, float* __restrict__ aw) {
  __shared__ float q[C_];
  const int row = blockIdx.x;
  const int t   = threadIdx.x;
  for (int i = t; i < C_; i += 96) q[i] = qn[(size_t)row * C_ + i];
  __syncthreads();
  if (t < NOFF) {
    float s = boff[t];
    #pragma unroll 4
    for (int k = 0; k < C_; ++k) s += q[k] * Woff[(size_t)k * NOFF + t];
    off[(size_t)row * NOFF + t] = s;
  } else if (t < NOFF + NAW) {
    const int col = t - NOFF;
    float s = baw[col];
    #pragma unroll 4
    for (int k = 0; k < C_; ++k) s += q[k] * Waw[(size_t)k * NAW + col];
    aw[(size_t)row * NAW + col] = s;
  }
}

// ---------------- softmax over the P=4 points per (row, head) ----------------
__global__ __launch_bounds__(256)
void softmax_kernel(float* __restrict__ aw, int n) {
  int i = blockIdx.x * blockDim.x + threadIdx.x;
  if (i >= n) return;
  float* p = aw + (size_t)i * PP;
  float p0 = p[0], p1 = p[1], p2 = p[2], p3 = p[3];
  float m = fmaxf(fmaxf(p0, p1), fmaxf(p2, p3));
  float e0 = __expf(p0 - m), e1 = __expf(p1 - m), e2 = __expf(p2 - m), e3 = __expf(p3 - m);
  float inv = 1.0f / (e0 + e1 + e2 + e3);
  p[0] = e0 * inv; p[1] = e1 * inv; p[2] = e2 * inv; p[3] = e3 * inv;
}

// ---------------- bilinear sampling + point-weighted sum ----------------
// one wave per (b, q, h); lanes stripe Dh=64 as {lane, lane+32}
__global__ __launch_bounds__(256)
void sample_kernel(const float* __restrict__ value, const float* __restrict__ ref,
                   const float* __restrict__ off, const float* __restrict__ aw,
                   float* __restrict__ attn) {
  const int wave = (blockIdx.x * blockDim.x + threadIdx.x) >> 5;
  const int lane = threadIdx.x & 31;
  if (wave >= R_ * HH) return;
  const int h  = wave % HH;
  const int rq = wave / HH;          // rq = b*LQ + q
  const int b  = rq / LQ;

  const float rx = ref[(size_t)rq * 2 + 0];
  const float ry = ref[(size_t)rq * 2 + 1];
  const float* vbase = value + ((size_t)b * LQ) * C_ + h * DH;  // + pos*C_ per cell
  const float* offp  = off + (size_t)rq * NOFF + h * (PP * 2);
  const float* awp   = aw  + (size_t)rq * NAW  + h * PP;

  float acc0 = 0.f, acc1 = 0.f;
  #pragma unroll
  for (int p = 0; p < PP; ++p) {
    // offset normalizer is [W, H]
    const float lx = rx + offp[p * 2 + 0] * (1.0f / GW);
    const float ly = ry + offp[p * 2 + 1] * (1.0f / GH);
    const float x = lx * GW - 0.5f;
    const float y = ly * GH - 0.5f;
    const float xf = floorf(x), yf = floorf(y);
    const int   x0 = (int)xf,  y0 = (int)yf;
    const float wx = x - xf,   wy = y - yf;
    const float a  = awp[p];
    const float cw[4] = { (1.f - wx) * (1.f - wy) * a, wx * (1.f - wy) * a,
                          (1.f - wx) * wy * a,         wx * wy * a };
    const int cx[4] = { x0, x0 + 1, x0,     x0 + 1 };
    const int cy[4] = { y0, y0,     y0 + 1, y0 + 1 };
    #pragma unroll
    for (int cidx = 0; cidx < 4; ++cidx) {
      const int xi = cx[cidx], yi = cy[cidx];
      if (xi >= 0 && xi < GW && yi >= 0 && yi < GH) {
        const float* v = vbase + (size_t)(yi * GW + xi) * C_;
        acc0 += cw[cidx] * v[lane];
        acc1 += cw[cidx] * v[lane + 32];
      }
    }
  }
  float* o = attn + (size_t)rq * C_ + h * DH;
  o[lane] = acc0;
  o[lane + 32] = acc1;
}

// ---------------- driver ----------------
extern "C" void kernel_launch(void* const* d_in, const int* in_sizes, int n_in,
                              void* d_out, int out_size, void* d_ws, size_t ws_size,
                              hipStream_t stream) {
  const float* query      = (const float*)d_in[0];
  const float* refpts     = (const float*)d_in[1];
  const float* feat       = (const float*)d_in[2];
  const float* feat_other = (const float*)d_in[3];
  const float* qn_g       = (const float*)d_in[4];
  const float* qn_b       = (const float*)d_in[5];
  const float* fn_g       = (const float*)d_in[6];
  const float* fn_b       = (const float*)d_in[7];
  const float* Wv         = (const float*)d_in[8];
  const float* bv         = (const float*)d_in[9];
  const float* Woff       = (const float*)d_in[10];
  const float* boff       = (const float*)d_in[11];
  const float* Waw        = (const float*)d_in[12];
  const float* baw        = (const float*)d_in[13];
  const float* Wout       = (const float*)d_in[14];
  const float* bout       = (const float*)d_in[15];
  const float* gamma      = (const float*)d_in[16];
  (void)in_sizes; (void)n_in; (void)out_size; (void)ws_size;

  float* out = (float*)d_out;
  float* ws  = (float*)d_ws;
  const size_t RC = (size_t)R_ * C_;
  float* qn    = ws;              // RC
  float* fn    = qn    + RC;      // RC
  float* value = fn    + RC;      // RC
  float* attn  = value + RC;      // RC
  float* offb  = attn  + RC;      // R_*48
  float* awb   = offb  + (size_t)R_ * NOFF;  // R_*24

  // qn = LN(query); out = query
  ln_kernel<<<R_, 128, 0, stream>>>(query, qn_g, qn_b, qn);
  copy_kernel<<<(int)(RC / 4 + 255) / 256, 256, 0, stream>>>(
      (const float4*)query, (float4*)out, (int)(RC / 4));

  const dim3 g16(R_ / 128, C_ / 16);  // 64 x 24 macro-tiles, 8 waves each
  for (int m = 0; m < MM; ++m) {
    const float* f = (m == 0) ? feat : (feat_other + (size_t)(m - 1) * RC);
    ln_kernel<<<R_, 128, 0, stream>>>(f, fn_g + m * C_, fn_b + m * C_, fn);
    // value = fn @ Wv[m] + bv[m]
    gemm_wmma_f32<false><<<g16, 256, 0, stream>>>(fn, Wv + (size_t)m * C_ * C_,
                                                  bv + m * C_, value, nullptr, nullptr);
    // off / aw projections from qn, then softmax over points
    offaw_kernel<<<R_, 96, 0, stream>>>(qn, Woff + (size_t)m * C_ * NOFF,
                                        boff + m * NOFF,
                                        Waw + (size_t)m * C_ * NAW,
                                        baw + m * NAW, offb, awb);
    softmax_kernel<<<(R_ * HH + 255) / 256, 256, 0, stream>>>(awb, R_ * HH);
    // bilinear sample + point-weighted reduce -> attn (B,Lq,H,Dh)
    sample_kernel<<<(R_ * HH * 32) / 256, 256, 0, stream>>>(value, refpts, offb, awb, attn);
    // out += gamma[m] * (attn @ Wout[m] + bout[m])
    gemm_wmma_f32<true><<<g16, 256, 0, stream>>>(attn, Wout + (size_t)m * C_ * C_,
                                                 bout + m * C_, nullptr,
                                                 gamma + m * C_, out);
  }
}